// ADRC_PE_21827023798438
// MI455X (gfx1250) — compile-verified
//
#include <hip/hip_runtime.h>
#include <hip/hip_bf16.h>
#include <math.h>

// ---------------------------------------------------------------------------
// Problem constants (from the reference)
// ---------------------------------------------------------------------------
#define BATCH   8
#define CIN     256
#define CRED    64
#define HH      160
#define WW      160
#define HWPIX   (HH * WW)                    // 25600
#define NPIX    (BATCH * HWPIX)              // 204800 flat (b,h,w) pixels
#define EPSV    0.0001f

typedef __attribute__((ext_vector_type(2))) float v2f;
typedef __attribute__((ext_vector_type(4))) float v4f;
typedef __attribute__((ext_vector_type(8))) float v8f;

__device__ __forceinline__ v4f vabs4(v4f a) {
    a.x = __builtin_fabsf(a.x); a.y = __builtin_fabsf(a.y);
    a.z = __builtin_fabsf(a.z); a.w = __builtin_fabsf(a.w);
    return a;
}

// ---------------------------------------------------------------------------
// Pass 1: y[p][d] = sum_c x[b,c,h,w] * reduce_w[d,c]   via V_WMMA_F32_16X16X4
// Also accumulates per-(b,d) pool sums (LDS atomics -> 1 global atomic/block).
//
// Block: 256 threads = 8 waves. Each wave: 16 pixels x 64 out-channels
// (4 WMMA accum tiles), K loop over 256 in steps of 4.
//
// Dynamic LDS layout:
//   [0 .. 16384)  float2 wP[kp][n] = { W[2kp][n], W[2kp+1][n] }  (kp=0..127)
//                 -> B fragment = ONE aligned ds_load_b64, no repack movs
//   [16384 .. +64) per-channel pool accumulator
// ---------------------------------------------------------------------------
#define K_STEP(k0, PREFETCH)                                                   \
    do {                                                                       \
        v2f A;                                                                 \
        A.x = xa[(size_t)(k0)*HWPIX];                                          \
        A.y = xa[(size_t)((k0) + 1) * HWPIX];                                  \
        if (PREFETCH)                                                          \
            __builtin_prefetch(xa + (size_t)((k0) + 16) * HWPIX, 0, 1);        \
        const v2f* wrow = wP + (((k0) >> 1) + ks) * CRED + m;                  \
        const v2f B0 = wrow[0];                                                \
        const v2f B1 = wrow[16];                                               \
        const v2f B2 = wrow[32];                                               \
        const v2f B3 = wrow[48];                                               \
        acc0 = __builtin_amdgcn_wmma_f32_16x16x4_f32(false, A, false, B0,      \
                                                     (short)0, acc0, false, false); \
        acc1 = __builtin_amdgcn_wmma_f32_16x16x4_f32(false, A, false, B1,      \
                                                     (short)0, acc1, false, false); \
        acc2 = __builtin_amdgcn_wmma_f32_16x16x4_f32(false, A, false, B2,      \
                                                     (short)0, acc2, false, false); \
        acc3 = __builtin_amdgcn_wmma_f32_16x16x4_f32(false, A, false, B3,      \
                                                     (short)0, acc3, false, false); \
    } while (0)

__global__ void adrc_reduce_wmma(const float* __restrict__ x,
                                 const float* __restrict__ reduce_w,
                                 float* __restrict__ y_t,        // [NPIX][64]
                                 float* __restrict__ pool_sum)   // [8][64]
{
    extern __shared__ float sm1[];
    v2f*   wP    = (v2f*)sm1;              // wP[kp*64 + n]
    float* lpool = sm1 + CIN * CRED;

    const int tid = threadIdx.x;

    // Stage paired weights: wP[kp][n] = {reduce_w[n][2kp], reduce_w[n][2kp+1]}.
    // n fastest -> contiguous 8B LDS writes across lanes (conflict-free).
    for (int i = tid; i < (CIN / 2) * CRED; i += 256) {
        const int n  = i & (CRED - 1);
        const int kp = i >> 6;
        wP[kp * CRED + n] = *(const v2f*)(reduce_w + n * CIN + 2 * kp);
    }
    if (tid < CRED) lpool[tid] = 0.0f;
    __syncthreads();

    const int wave  = tid >> 5;            // 0..7 (wave32)
    const int lane  = tid & 31;
    const int m     = lane & 15;           // row within 16-pixel tile
    const int ks    = lane >> 4;           // half-wave selects K pair {0,1}/{2,3}

    const int pbase = blockIdx.x * 128 + wave * 16;   // flat pixel base
    const int b     = pbase / HWPIX;                  // 128 | 25600 -> uniform b
    const int hwb   = pbase - b * HWPIX;

    // Per-lane base pointer: x[b, (2*ks), hwb + m]
    const float* xa = x + (size_t)b * CIN * HWPIX + (size_t)(2 * ks) * HWPIX
                        + (size_t)(hwb + m);

    v8f acc0 = {}, acc1 = {}, acc2 = {}, acc3 = {};

    // Main body: guard-free, unconditional prefetch (stays in-bounds: k<256).
    for (int k0 = 0; k0 < CIN - 16; k0 += 4) {
        K_STEP(k0, 1);
    }
    // Tail: last 4 k-steps, no prefetch (would run past the channel range).
    for (int k0 = CIN - 16; k0 < CIN; k0 += 4) {
        K_STEP(k0, 0);
    }

    // D layout: lane gives N = m (+ tile*16); VGPR r gives M = r + 8*ks.
    // Store y pixel-major: y_t[(pbase + M) * 64 + N]  (coalesced 64B runs).
    float* yt = y_t + (size_t)pbase * CRED;
    v8f accs[4] = {acc0, acc1, acc2, acc3};
#pragma unroll
    for (int t = 0; t < 4; ++t) {
        const int n = t * 16 + m;
        float psum = 0.0f;
#pragma unroll
        for (int r = 0; r < 8; ++r) {
            const int pl = r + 8 * ks;
            float v = accs[t][r];
            yt[pl * CRED + n] = v;
            psum += v;
        }
        atomicAdd(&lpool[n], psum);   // ds_add_f32
    }
    __syncthreads();
    if (tid < CRED) atomicAdd(&pool_sum[b * CRED + tid], lpool[tid]);
}

// ---------------------------------------------------------------------------
// Pass 2: tiny gate MLP.  w_eff[b][d] = fw_a[d] + sigmoid(h@W2+b2)[b][d]*fw_b[d]
// ---------------------------------------------------------------------------
__global__ void adrc_gate(const float* __restrict__ pool_sum,
                          const float* __restrict__ gw1,  // [16][64]
                          const float* __restrict__ gb1,  // [16]
                          const float* __restrict__ gw2,  // [64][16]
                          const float* __restrict__ gb2,  // [64]
                          const float* __restrict__ fw,   // [128]
                          float* __restrict__ weff)       // [8][64]
{
    __shared__ float pooled_s[BATCH * CRED];
    __shared__ float h_s[BATCH * 16];
    const int tid = threadIdx.x;

    for (int i = tid; i < BATCH * CRED; i += 256)
        pooled_s[i] = pool_sum[i] * (1.0f / (float)HWPIX);
    __syncthreads();

    if (tid < BATCH * 16) {
        const int b = tid >> 4, j = tid & 15;
        float acc = gb1[j];
        for (int d = 0; d < CRED; ++d)
            acc += pooled_s[b * CRED + d] * gw1[j * CRED + d];
        h_s[tid] = fmaxf(acc, 0.0f);
    }
    __syncthreads();

    for (int i = tid; i < BATCH * CRED; i += 256) {
        const int b = i >> 6, d = i & 63;
        float acc = gb2[d];
        for (int j = 0; j < 16; ++j)
            acc += h_s[b * 16 + j] * gw2[d * 16 + j];
        const float g = 1.0f / (1.0f + __expf(-acc));
        weff[i] = fw[d] + g * fw[CRED + d];
    }
}

// ---------------------------------------------------------------------------
// Pass 3: per 32x8 spatial tile (one per block, one pixel per thread):
//  - stage 34x10 halo x 64 channels of y into LDS (padded stride 68)
//  - 3x3 stencils -> kappa -> logits = dot(kappa, w_eff[b])
//  - scale = 1 + sigmoid(logits); out[b,c,h,w] = x * scale for all 256 c.
// ---------------------------------------------------------------------------
#define TW   32
#define TH   8
#define HALW (TW + 2)     // 34
#define HALH (TH + 2)     // 10
#define CPAD 68           // 64 channels padded -> conflict-free b128 reads

__global__ void adrc_fuse(const float* __restrict__ x,
                          const float* __restrict__ y_t,   // [NPIX][64]
                          const float* __restrict__ weff,  // [8][64]
                          float* __restrict__ outp)
{
    extern __shared__ float sm[];            // HALW*HALH*CPAD floats
    const int tid = threadIdx.x;

    const int blk = blockIdx.x;              // 8 * 5 * 20 = 800 blocks
    const int b   = blk / 100;
    const int t   = blk - b * 100;
    const int ty  = t / 5;                   // 0..19
    const int tx  = t - ty * 5;              // 0..4
    const int h0  = ty * TH;
    const int w0  = tx * TW;

    // ---- stage halo tile (zero-padded at image borders), float4 chunks ----
    for (int i = tid; i < HALW * HALH * 16; i += 256) {
        const int pix = i >> 4;              // 0..339
        const int j   = i & 15;              // float4 index within 64 channels
        const int py  = pix / HALW;
        const int px  = pix - py * HALW;
        const int h   = h0 + py - 1;
        const int w   = w0 + px - 1;
        v4f v = {};
        if (h >= 0 && h < HH && w >= 0 && w < WW) {
            const size_t off = ((size_t)(b * HWPIX + h * WW + w)) * CRED + j * 4;
            v = *(const v4f*)(y_t + off);
        }
        *(v4f*)&sm[pix * CPAD + j * 4] = v;
    }
    __syncthreads();

    // ---- stencil + logits ----
    const int txl = tid & 31;
    const int tyl = tid >> 5;
    const v4f* wv = (const v4f*)(weff + b * CRED);

    float logits = 0.0f;
#pragma unroll 4
    for (int d4 = 0; d4 < 16; ++d4) {
        v4f n00, n01, n02, n10, n11, n12, n20, n21, n22;
        {
            const float* base = sm + d4 * 4;
            n00 = *(const v4f*)&base[((tyl + 0) * HALW + txl + 0) * CPAD];
            n01 = *(const v4f*)&base[((tyl + 0) * HALW + txl + 1) * CPAD];
            n02 = *(const v4f*)&base[((tyl + 0) * HALW + txl + 2) * CPAD];
            n10 = *(const v4f*)&base[((tyl + 1) * HALW + txl + 0) * CPAD];
            n11 = *(const v4f*)&base[((tyl + 1) * HALW + txl + 1) * CPAD];
            n12 = *(const v4f*)&base[((tyl + 1) * HALW + txl + 2) * CPAD];
            n20 = *(const v4f*)&base[((tyl + 2) * HALW + txl + 0) * CPAD];
            n21 = *(const v4f*)&base[((tyl + 2) * HALW + txl + 1) * CPAD];
            n22 = *(const v4f*)&base[((tyl + 2) * HALW + txl + 2) * CPAD];
        }
        const v4f mu = (n00 + n01 + n02 + n10 + n11 + n12 + n20 + n21 + n22)
                       * (1.0f / 9.0f);
        const v4f gx = (n00 - n02 + 2.0f * (n10 - n12) + n20 - n22) * 0.125f;
        const v4f gy = (n00 + 2.0f * n01 + n02 - n20 - 2.0f * n21 - n22) * 0.125f;
        const v4f gm = vabs4(gx) + vabs4(gy);
        const v4f kp = 1.0f - vabs4(n11 - mu) / (gm + EPSV);
        const v4f wq = wv[d4];
        logits += kp.x * wq.x + kp.y * wq.y + kp.z * wq.z + kp.w * wq.w;
    }

    const float a     = 1.0f / (1.0f + __expf(-logits));
    const float scale = 1.0f + a;

    __syncthreads();                 // tile reads done; reuse LDS for scales
    sm[tyl * TW + txl] = scale;      // sm[0..255]
    __syncthreads();

    // ---- out = x * scale over all 256 channels, float4 along w ----
    for (int i = tid; i < CIN * (TW * TH / 4); i += 256) {
        const int c  = i >> 6;               // 0..255
        const int r  = i & 63;               // float4 slot in tile
        const int py = r >> 3;
        const int px = (r & 7) << 2;
        const size_t off = ((size_t)(b * CIN + c)) * HWPIX
                         + (size_t)(h0 + py) * WW + (w0 + px);
        const v4f xv = *(const v4f*)(x + off);
        const v4f sc = *(const v4f*)&sm[py * TW + px];
        *(v4f*)(outp + off) = xv * sc;
    }
}

// ---------------------------------------------------------------------------
// Launch
// ---------------------------------------------------------------------------
extern "C" void kernel_launch(void* const* d_in, const int* in_sizes, int n_in,
                              void* d_out, int out_size, void* d_ws, size_t ws_size,
                              hipStream_t stream) {
    const float* x        = (const float*)d_in[0];
    const float* reduce_w = (const float*)d_in[1];
    const float* gw1      = (const float*)d_in[2];
    const float* gb1      = (const float*)d_in[3];
    const float* gw2      = (const float*)d_in[4];
    const float* gb2      = (const float*)d_in[5];
    const float* fw       = (const float*)d_in[6];
    float* outp           = (float*)d_out;

    // Workspace: y_t [NPIX*64] f32 (52.4 MB), then pool sums [512], weff [512].
    float* y_t      = (float*)d_ws;
    float* pool_sum = (float*)((char*)d_ws + (size_t)NPIX * CRED * sizeof(float));
    float* weff     = pool_sum + BATCH * CRED;

    hipMemsetAsync(pool_sum, 0, BATCH * CRED * sizeof(float), stream);

    const size_t lds1 = (CIN * CRED + CRED) * sizeof(float);        // 65,792 B
    adrc_reduce_wmma<<<NPIX / 128, 256, lds1, stream>>>(x, reduce_w, y_t, pool_sum);

    adrc_gate<<<1, 256, 0, stream>>>(pool_sum, gw1, gb1, gw2, gb2, fw, weff);

    const size_t lds3 = (size_t)HALW * HALH * CPAD * sizeof(float); // 92,480 B
    adrc_fuse<<<BATCH * (HH / TH) * (WW / TW), 256, lds3, stream>>>(x, y_t, weff, outp);
}